// Encoder_2379411882186
// MI455X (gfx1250) — compile-verified
//
#include <hip/hip_runtime.h>
#include <hip/hip_bf16.h>

typedef __attribute__((ext_vector_type(2))) float v2f;
typedef __attribute__((ext_vector_type(8))) float v8f;

#define N_NODES 50000
#define N_EDGES 600000
#define DIM     128
#define NGRAPH  64
#define DEPTH   4

// ---------------------------------------------------------------------------
// Utility: zero a float buffer
__global__ void zero_f32_kernel(float* __restrict__ p, int n) {
    int i = blockIdx.x * blockDim.x + threadIdx.x;
    if (i < n) p[i] = 0.0f;
}

// In-degree accumulation: deg[dst[e]] += 1
__global__ void deg_kernel(const int* __restrict__ dst, float* __restrict__ deg) {
    int e = blockIdx.x * blockDim.x + threadIdx.x;
    if (e < N_EDGES) atomicAdd(&deg[dst[e]], 1.0f);
}

// dis[n] = deg>0 ? rsqrt(deg) : 0
__global__ void dis_kernel(const float* __restrict__ deg, float* __restrict__ dis) {
    int i = blockIdx.x * blockDim.x + threadIdx.x;
    if (i < N_NODES) {
        float d = deg[i];
        dis[i] = (d > 0.0f) ? rsqrtf(d) : 0.0f;
    }
}

// Embedding lookup: h[n, :] = embedding[x[n], :]  (float4 granularity)
__global__ void embed_kernel(const int* __restrict__ x,
                             const float* __restrict__ emb,
                             float* __restrict__ h) {
    int i = blockIdx.x * blockDim.x + threadIdx.x;   // one float4 each
    if (i >= N_NODES * (DIM / 4)) return;
    int node = i >> 5;          // DIM/4 == 32
    int d4   = i & 31;
    int tok  = x[node];
    ((float4*)h)[i] = ((const float4*)(emb + (long)tok * DIM))[d4];
}

// h[n, d] = bias[d]  (scatter target re-init, float4 granularity)
__global__ void bias_init_kernel(const float* __restrict__ bias, float* __restrict__ h) {
    int i = blockIdx.x * blockDim.x + threadIdx.x;
    if (i >= N_NODES * (DIM / 4)) return;
    int d4 = i & 31;
    ((float4*)h)[i] = ((const float4*)bias)[d4];
}

// ---------------------------------------------------------------------------
// hw = h @ W using V_WMMA_F32_16X16X4_F32.
// Grid: 3125 blocks (50000/16 row tiles) x 256 threads (8 waves).
// Each wave computes one 16x16 output tile; wave id selects column tile (DIM/16=8).
// The 16x128 A-strip is staged in LDS once per block and shared by all 8 waves.
__global__ void __launch_bounds__(256)
gemm_wmma_kernel(const float* __restrict__ h,
                 const float* __restrict__ W,
                 float* __restrict__ hw) {
    __shared__ float As[16 * DIM];   // 8 KB

    const int tid     = threadIdx.x;
    const int rowBase = blockIdx.x * 16;

    // Cooperative load of the 16x128 strip (2048 floats = 512 float4, 256 thr -> 2 each)
    {
        const float4* src = (const float4*)(h + (long)rowBase * DIM);
        float4* dst4 = (float4*)As;
        dst4[tid]       = src[tid];
        dst4[tid + 256] = src[tid + 256];
    }
    __syncthreads();

    const int wave    = tid >> 5;      // 0..7 -> column tile
    const int lane    = tid & 31;
    const int laneM   = lane & 15;
    const int half    = lane >> 4;     // 0 or 1 (selects K=+0/+1 vs K=+2/+3)
    const int colBase = wave * 16;

    v8f acc = {};                       // C = 0 accumulator

    #pragma unroll
    for (int k0 = 0; k0 < DIM; k0 += 4) {
        const int kA = k0 + half * 2;
        // A-matrix 16x4 f32 layout: lanes 0-15 rows M, VGPR0=K+0 VGPR1=K+1;
        //                           lanes 16-31 rows M, VGPR0=K+2 VGPR1=K+3.
        v2f a;
        a.x = As[laneM * DIM + kA];
        a.y = As[laneM * DIM + kA + 1];
        // B-matrix 4x16 f32: row (K) striped across lanes (N=laneM), mirrored halves.
        v2f b;
        b.x = W[kA * DIM + colBase + laneM];
        b.y = W[(kA + 1) * DIM + colBase + laneM];
        // 8 args: (neg_a, A, neg_b, B, c_mod, C, reuse_a, reuse_b)
        acc = __builtin_amdgcn_wmma_f32_16x16x4_f32(
            false, a, false, b, (short)0, acc, false, false);
    }

    // C/D layout: VGPR r -> lanes 0-15: M=r, lanes 16-31: M=r+8; N = laneM
    float* out = hw + (long)rowBase * DIM + colBase;
    #pragma unroll
    for (int r = 0; r < 8; ++r) {
        out[(r + half * 8) * DIM + laneM] = acc[r];
    }
}

// ---------------------------------------------------------------------------
// Edge scatter: h[dst] += hw[src] * (dis[src]*dis[dst]).
// One wave per edge; each lane handles 4 consecutive floats of the 128-dim row.
__global__ void __launch_bounds__(256)
scatter_kernel(const float* __restrict__ hw,
               const int* __restrict__ src_idx,
               const int* __restrict__ dst_idx,
               const float* __restrict__ dis,
               float* __restrict__ h) {
    int gtid = blockIdx.x * blockDim.x + threadIdx.x;
    int edge = gtid >> 5;
    int lane = threadIdx.x & 31;
    if (edge >= N_EDGES) return;

    int s = src_idx[edge];
    int d = dst_idx[edge];
    float nrm = dis[s] * dis[d];

    float4 v = ((const float4*)(hw + (long)s * DIM))[lane];
    float* out = h + (long)d * DIM + lane * 4;
    atomicAdd(out + 0, v.x * nrm);
    atomicAdd(out + 1, v.y * nrm);
    atomicAdd(out + 2, v.z * nrm);
    atomicAdd(out + 3, v.w * nrm);
}

// Final gather: out[b, :] = h[ptr[b+1]-1, :]
__global__ void gather_out_kernel(const int* __restrict__ ptr,
                                  const float* __restrict__ h,
                                  float* __restrict__ out) {
    int i = blockIdx.x * blockDim.x + threadIdx.x;
    if (i >= NGRAPH * DIM) return;
    int b = i >> 7;
    int d = i & (DIM - 1);
    int node = ptr[b + 1] - 1;
    out[i] = h[(long)node * DIM + d];
}

// ---------------------------------------------------------------------------
extern "C" void kernel_launch(void* const* d_in, const int* in_sizes, int n_in,
                              void* d_out, int out_size, void* d_ws, size_t ws_size,
                              hipStream_t stream) {
    const int*   x    = (const int*)d_in[0];
    const int*   eidx = (const int*)d_in[1];   // [2, E]: src then dst
    const int*   ptr  = (const int*)d_in[2];
    /* d_in[3] = depth (==4, hardcoded as DEPTH) */
    const float* emb  = (const float*)d_in[4];
    const float* W    = (const float*)d_in[5];
    const float* bias = (const float*)d_in[6];
    float*       out  = (float*)d_out;

    const int* src_idx = eidx;
    const int* dst_idx = eidx + N_EDGES;

    // Workspace partition
    float* h   = (float*)d_ws;
    float* hw  = h  + (size_t)N_NODES * DIM;
    float* deg = hw + (size_t)N_NODES * DIM;
    float* dis = deg + N_NODES;

    const int T = 256;

    // Degrees + normalization factors
    zero_f32_kernel<<<(N_NODES + T - 1) / T, T, 0, stream>>>(deg, N_NODES);
    deg_kernel<<<(N_EDGES + T - 1) / T, T, 0, stream>>>(dst_idx, deg);
    dis_kernel<<<(N_NODES + T - 1) / T, T, 0, stream>>>(deg, dis);

    // Embedding lookup
    const int vec_elems = N_NODES * (DIM / 4);
    embed_kernel<<<(vec_elems + T - 1) / T, T, 0, stream>>>(x, emb, h);

    // GCN layers
    for (int l = 0; l < DEPTH; ++l) {
        gemm_wmma_kernel<<<N_NODES / 16, T, 0, stream>>>(h, W, hw);
        bias_init_kernel<<<(vec_elems + T - 1) / T, T, 0, stream>>>(bias, h);
        scatter_kernel<<<(N_EDGES * 32 + T - 1) / T, T, 0, stream>>>(hw, src_idx, dst_idx, dis, h);
    }

    // Output gather
    gather_out_kernel<<<(NGRAPH * DIM + T - 1) / T, T, 0, stream>>>(ptr, h, out);
}